// TimeStretch_72739566125443
// MI455X (gfx1250) — compile-verified
//
#include <hip/hip_runtime.h>
#include <math.h>

// ---------------- problem constants (from the reference) ----------------
#define BC      32          // B*C = 16*2
#define L_IN    262144
#define NFFT    2048
#define HOP     512
#define PADH    1024        // n_fft/2 reflect pad
#define T_IN    513         // 1 + (L + 2*PADH - NFFT)/HOP
#define NFREQ   1025        // n_fft/2 + 1
#define T_OUT   411         // len(np.arange(0, 513, 1.25))
#define OUT_LEN 209715      // int(262144 / 1.25)
#define KPAD    1028        // NFREQ padded to multiple of 4
#define KCH     128         // ISTFT K-chunk staged in LDS

#define TWO_PI_F 6.28318530717958647692f
#define PI_F     3.14159265358979323846f

typedef float v2f __attribute__((ext_vector_type(2)));
typedef float v8f __attribute__((ext_vector_type(8)));

// =======================================================================
// K1: STFT as real-DFT GEMM on the fp32 WMMA pipe.
//     M = 16 freq bins, N = 16 frames, K = 2048 samples (steps of 4).
//     Re[f] = sum x[n] cos(2*pi*f*n/N) ; Im[f] = -sum x[n] sin(...)
//     Stores magnitude + angle (what the phase vocoder consumes).
// =======================================================================
__global__ __launch_bounds__(256) void stft_wmma(const float* __restrict__ x,
                                                 float* __restrict__ specMag,
                                                 float* __restrict__ specAng) {
  __shared__ float lutc[NFFT];
  __shared__ float luts[NFFT];
  __shared__ float fr[15 * HOP + NFFT];              // 9728 floats: 16 overlapping frames

  const int tid   = threadIdx.x;
  const int ttile = blockIdx.x;                      // 0..32 (33 tiles of 16 frames)
  const int bc    = blockIdx.y;

  for (int i = tid; i < NFFT; i += 256) {
    float a = (TWO_PI_F / (float)NFFT) * (float)i;
    lutc[i] = __cosf(a);
    luts[i] = __sinf(a);
  }
  const float* xb  = x + (size_t)bc * L_IN;
  const int    base = ttile * 16 * HOP;              // offset into reflect-padded stream
  for (int i = tid; i < 15 * HOP + NFFT; i += 256) {
    int j = base + i - PADH;                         // index into original signal
    if (j < 0)      j = -j;                          // reflect left
    if (j >= L_IN)  j = 2 * L_IN - 2 - j;            // reflect right
    fr[i] = xb[j];
  }
  __syncthreads();

  const int wave = tid >> 5;
  const int lane = tid & 31;
  const int lh   = lane >> 4;                        // half-wave select (K offset / row+8)
  const int lm   = lane & 15;
  const int t    = ttile * 16 + lm;                  // output frame (N = lane)

  for (int ft = wave; ft < 65; ft += 8) {            // 65 tiles cover 1040 >= 1025 bins
    v8f accR = {};
    v8f accI = {};
    const int fA = ft * 16 + lm;                     // A row = freq bin for this lane
    for (int k0 = 0; k0 < NFFT; k0 += 4) {
      const int ka = k0 + 2 * lh;                    // A/B K layout: lanes0-15 K0..1, 16-31 K2..3
      const int i0 = (fA * ka) & (NFFT - 1);
      const int i1 = (fA * (ka + 1)) & (NFFT - 1);
      v2f aR, aI, b;
      aR.x = lutc[i0];  aR.y = lutc[i1];
      aI.x = -luts[i0]; aI.y = -luts[i1];
      b.x  = fr[lm * HOP + ka];
      b.y  = fr[lm * HOP + ka + 1];
      accR = __builtin_amdgcn_wmma_f32_16x16x4_f32(false, aR, false, b, (short)0, accR, false, false);
      accI = __builtin_amdgcn_wmma_f32_16x16x4_f32(false, aI, false, b, (short)0, accI, false, false);
    }
    if (t < T_IN) {
#pragma unroll
      for (int v = 0; v < 8; ++v) {
        const int f = ft * 16 + v + 8 * lh;          // C/D layout: VGPR v = rows v, v+8
        if (f < NFREQ) {
          const float re = accR[v], im = accI[v];
          const size_t o = ((size_t)bc * NFREQ + f) * T_IN + t;
          specMag[o] = sqrtf(re * re + im * im);
          specAng[o] = atan2f(im, re);
        }
      }
    }
  }
}

// =======================================================================
// K2: phase vocoder. One lane per (bc, freq) row; serial phase cumsum over
//     the 411 output steps. Folds the irfft weights (1/N, 2/N, -Im) into
//     the stretched spectrum so K3 is a plain  cos*bR + sin*bI  GEMM.
// =======================================================================
__global__ __launch_bounds__(256) void phase_vocoder(const float* __restrict__ specMag,
                                                     const float* __restrict__ specAng,
                                                     float* __restrict__ strRe,
                                                     float* __restrict__ strIm) {
  const int idx = blockIdx.x * blockDim.x + threadIdx.x;
  if (idx >= BC * NFREQ) return;
  const int bc = idx / NFREQ;
  const int f  = idx - bc * NFREQ;

  const float dphi = PI_F * (float)HOP * ((float)f / (float)(NFREQ - 1)); // linspace(0, pi*hop)
  const size_t row = (size_t)bc * NFREQ + f;
  const float* mg = specMag + row * T_IN;
  const float* an = specAng + row * T_IN;
  float* oR = strRe + row * T_OUT;
  float* oI = strIm + row * T_OUT;

  const float w = ((f == 0) || (f == NFREQ - 1)) ? (1.0f / (float)NFFT) : (2.0f / (float)NFFT);

  float acc = an[0];                                  // phase_0
  float pa0 = 0.0f, pa1 = 0.0f;
  for (int j = 0; j < T_OUT; ++j) {
    const float ts   = 1.25f * (float)j;              // exact in binary
    const int   i0   = (int)ts;                       // <= 512 always
    const float alpha = ts - (float)i0;
    const float n0 = mg[i0], a0 = an[i0];
    float n1, a1;
    if (i0 + 1 < T_IN) { n1 = mg[i0 + 1]; a1 = an[i0 + 1]; }
    else               { n1 = 0.0f;       a1 = 0.0f; }      // zero-pad column
    if (j > 0) {                                      // ph shifted by one step
      float ph = pa1 - pa0 - dphi;
      ph -= TWO_PI_F * rintf(ph * (1.0f / TWO_PI_F)); // round-half-even wrap, like jnp.round
      acc += ph + dphi;
    }
    const float m = w * (alpha * n1 + (1.0f - alpha) * n0);
    oR[j] =  m * __cosf(acc);                         //  w * Re
    oI[j] = -m * __sinf(acc);                         // -w * Im (irfft sign folded in)
    pa0 = a0; pa1 = a1;
  }
}

// =======================================================================
// K3: ISTFT as GEMM: y[n,t] = sum_f cos(2pi n f/N)*bR[f,t] + sin(...)*bI[f,t]
//     M = 16 samples, N = 16 frames, K = 1028 bins. B staged in LDS in
//     128-wide K chunks (32KB LDS incl. LUT), shared by all 8 waves.
// =======================================================================
__global__ __launch_bounds__(256) void istft_wmma(const float* __restrict__ strRe,
                                                  const float* __restrict__ strIm,
                                                  float* __restrict__ frames) {
  __shared__ float lutc[NFFT];
  __shared__ float luts[NFFT];
  __shared__ float bR[KCH * 16];
  __shared__ float bI[KCH * 16];

  const int tid   = threadIdx.x;
  const int ttile = blockIdx.x;                       // 0..25 (26 tiles of 16 out-frames)
  const int bc    = blockIdx.y;

  for (int i = tid; i < NFFT; i += 256) {
    float a = (TWO_PI_F / (float)NFFT) * (float)i;
    lutc[i] = __cosf(a);
    luts[i] = __sinf(a);
  }
  __syncthreads();

  const int wave = tid >> 5;
  const int lane = tid & 31;
  const int lh   = lane >> 4;
  const int lm   = lane & 15;
  const int t    = ttile * 16 + lm;

  for (int pass = 0; pass < 16; ++pass) {             // 128 n-tiles / 8 waves
    const int nt = wave + pass * 8;
    const int nA = nt * 16 + lm;                      // A row = output sample index
    v8f acc = {};
    for (int base = 0; base < KPAD; base += KCH) {
      const int ch = (KPAD - base < KCH) ? (KPAD - base) : KCH;
      for (int i = tid; i < ch * 16; i += 256) {      // stage B chunk
        const int fg = base + (i >> 4);
        const int tc = ttile * 16 + (i & 15);
        float vr = 0.0f, vi = 0.0f;
        if (fg < NFREQ && tc < T_OUT) {
          const size_t o = ((size_t)bc * NFREQ + fg) * T_OUT + tc;
          vr = strRe[o];
          vi = strIm[o];
        }
        bR[i] = vr;
        bI[i] = vi;
      }
      __syncthreads();
      for (int k0 = 0; k0 < ch; k0 += 4) {
        const int fg = base + k0 + 2 * lh;
        const int i0 = (nA * fg) & (NFFT - 1);
        const int i1 = (nA * (fg + 1)) & (NFFT - 1);
        v2f aC, aS, b0, b1;
        aC.x = lutc[i0]; aC.y = lutc[i1];
        aS.x = luts[i0]; aS.y = luts[i1];
        const int kl = k0 + 2 * lh;
        b0.x = bR[kl * 16 + lm]; b0.y = bR[(kl + 1) * 16 + lm];
        b1.x = bI[kl * 16 + lm]; b1.y = bI[(kl + 1) * 16 + lm];
        acc = __builtin_amdgcn_wmma_f32_16x16x4_f32(false, aC, false, b0, (short)0, acc, false, false);
        acc = __builtin_amdgcn_wmma_f32_16x16x4_f32(false, aS, false, b1, (short)0, acc, false, false);
      }
      __syncthreads();
    }
    if (t < T_OUT) {
#pragma unroll
      for (int v = 0; v < 8; ++v) {
        const int n = nt * 16 + v + 8 * lh;
        frames[((size_t)bc * T_OUT + t) * NFFT + n] = acc[v];
      }
    }
  }
}

// =======================================================================
// K4: overlap-add in gather form (deterministic: no float atomics),
//     envelope = analytic count of contributing frames, offset n_fft/2,
//     zero-pad tail back to L (rate > 1 shortens the audio).
// =======================================================================
__global__ __launch_bounds__(256) void overlap_add(const float* __restrict__ frames,
                                                   float* __restrict__ out) {
  const size_t idx = (size_t)blockIdx.x * blockDim.x + threadIdx.x;
  if (idx >= (size_t)BC * L_IN) return;
  const int bc = (int)(idx / L_IN);
  const int s  = (int)(idx - (size_t)bc * L_IN);
  float r = 0.0f;
  if (s < OUT_LEN) {
    const int i = s + PADH;                           // index into the OLA stream
    int tmax = i >> 9;                                // i / HOP
    if (tmax > T_OUT - 1) tmax = T_OUT - 1;
    float sum = 0.0f;
    int cnt = 0;
    for (int tt = tmax; tt >= 0; --tt) {
      const int o = i - tt * HOP;
      if (o >= NFFT) break;
      sum += frames[((size_t)bc * T_OUT + tt) * NFFT + o];
      ++cnt;
    }
    r = sum / (float)cnt;
  }
  out[idx] = r;
}

// =======================================================================
// Workspace (floats):
//   specMag : BC*NFREQ*T_IN  = 16,826,400
//   specAng : BC*NFREQ*T_IN  = 16,826,400
//   strRe   : BC*NFREQ*T_OUT = 13,480,800
//   strIm   : BC*NFREQ*T_OUT = 13,480,800
//   frames  : BC*T_OUT*NFFT  = 26,935,296  (aliases spec region, dead by then)
// total = 60,614,400 floats ~= 242 MB
// =======================================================================
extern "C" void kernel_launch(void* const* d_in, const int* in_sizes, int n_in,
                              void* d_out, int out_size, void* d_ws, size_t ws_size,
                              hipStream_t stream) {
  const float* x = (const float*)d_in[0];
  float* out = (float*)d_out;
  float* w   = (float*)d_ws;

  const size_t SPEC = (size_t)BC * NFREQ * T_IN;
  const size_t STR  = (size_t)BC * NFREQ * T_OUT;

  float* specMag = w;
  float* specAng = w + SPEC;
  float* strRe   = w + 2 * SPEC;
  float* strIm   = w + 2 * SPEC + STR;
  float* frames  = w;                                 // reuse spec region

  stft_wmma<<<dim3(33, BC), 256, 0, stream>>>(x, specMag, specAng);
  phase_vocoder<<<(BC * NFREQ + 255) / 256, 256, 0, stream>>>(specMag, specAng, strRe, strIm);
  istft_wmma<<<dim3(26, BC), 256, 0, stream>>>(strRe, strIm, frames);
  overlap_add<<<(unsigned)(((size_t)BC * L_IN + 255) / 256), 256, 0, stream>>>(frames, out);
}